// SoftNormalShader_74294344286336
// MI455X (gfx1250) — compile-verified
//
#include <hip/hip_runtime.h>

// SoftNormalShader for MI455X (gfx1250, wave32).
//
// Roofline: ~218 MB streaming + L2-resident gathers (1.8 MB tables) ->
// ~9.4 us HBM floor at 23.3 TB/s; VALU trimmed (__expf, 32-bit index math)
// so compute stays under the memory floor. No WMMA applies (per-pixel
// 1x3 * 3x3 contractions, no shared operand matrix). CDNA5 paths used:
// global_load_async_to_lds_b128 staging (ASYNCcnt) + s_wait_asynccnt,
// global_prefetch for the gather tables, nontemporal b128 stream access.

typedef float f32x4 __attribute__((ext_vector_type(4)));
typedef int   i32x4 __attribute__((ext_vector_type(4)));

// GCC-vector int4 in explicit address spaces — matches the builtin's
// parameter types (v4i32 AS1 src / v4i32 AS3 dst).
typedef int v4i __attribute__((vector_size(16)));
typedef __attribute__((address_space(1))) v4i GAS1v4i;
typedef __attribute__((address_space(3))) v4i LAS3v4i;

#define HAS_ASYNC_LDS  __has_builtin(__builtin_amdgcn_global_load_async_to_lds_b128)
#define HAS_WAIT_ASYNC __has_builtin(__builtin_amdgcn_s_wait_asynccnt)

namespace {

constexpr int   KF        = 8;
constexpr int   TPB       = 256;        // threads per block == pixels per block
constexpr float SIGMA_INV = 1.0e4f;     // 1/1e-4
constexpr float GAMMA_INV = 1.0e4f;     // 1/1e-4
constexpr float ZFAR      = 100.0f;
constexpr float ZSCALE    = 1.0f / 99.0f;   // 1/(zfar-znear)
constexpr float EPSF      = 1.0e-10f;

__device__ __forceinline__ void stage16(const void* g, void* l) {
#if HAS_ASYNC_LDS
  // async global -> LDS, 16B per lane, tracked by ASYNCcnt
  __builtin_amdgcn_global_load_async_to_lds_b128((GAS1v4i*)g, (LAS3v4i*)l, 0, 0);
#else
  (void)g; (void)l;
#endif
}

__device__ __forceinline__ void wait_async0() {
#if HAS_WAIT_ASYNC
  __builtin_amdgcn_s_wait_asynccnt(0);
#else
  asm volatile("s_wait_asynccnt 0" ::: "memory");
#endif
}

struct PixIn {
  float bary[KF][3];
  int   pf[KF];
  float dist[KF];
  float z[KF];
};

__device__ __forceinline__ f32x4 shade(const PixIn& px,
                                       const float* __restrict__ vn,
                                       const int*   __restrict__ faces) {
  // Kick off gather-table prefetches before the ALU work.
#pragma unroll
  for (int k = 0; k < KF; ++k) {
    const int f = px.pf[k];
    if (f >= 0) __builtin_prefetch(faces + 3 * f, 0, 3);
  }

  float col[KF][3];
  float prob[KF], zinv[KF];
  float alpha = 1.0f, zmax = 0.0f;

#pragma unroll
  for (int k = 0; k < KF; ++k) {
    const int  f = px.pf[k];
    const bool m = (f >= 0);
    float c0 = 0.0f, c1 = 0.0f, c2 = 0.0f;
    if (m) {
      const int fb = 3 * f;                 // < 300000, fits i32
      const int i0 = 3 * faces[fb + 0];     // < 150000, fits i32
      const int i1 = 3 * faces[fb + 1];
      const int i2 = 3 * faces[fb + 2];
      const float b0 = px.bary[k][0], b1 = px.bary[k][1], b2 = px.bary[k][2];
      c0 = fmaf(b0, vn[i0 + 0], fmaf(b1, vn[i1 + 0], b2 * vn[i2 + 0]));
      c1 = fmaf(b0, vn[i0 + 1], fmaf(b1, vn[i1 + 1], b2 * vn[i2 + 1]));
      c2 = fmaf(b0, vn[i0 + 2], fmaf(b1, vn[i1 + 2], b2 * vn[i2 + 2]));
    }
    col[k][0] = c0; col[k][1] = c1; col[k][2] = c2;
    // sigmoid(-d/sigma) = 1/(1+exp(d/sigma)); saturates correctly via inf.
    const float pm = m ? 1.0f / (1.0f + __expf(px.dist[k] * SIGMA_INV)) : 0.0f;
    prob[k] = pm;
    alpha  *= (1.0f - pm);
    const float zi = m ? (ZFAR - px.z[k]) * ZSCALE : 0.0f;
    zinv[k] = zi;
    zmax    = fmaxf(zmax, zi);
  }

  float wsum = 0.0f, r = 0.0f, g = 0.0f, b = 0.0f;
#pragma unroll
  for (int k = 0; k < KF; ++k) {
    const float w = prob[k] * __expf((zinv[k] - zmax) * GAMMA_INV);
    wsum += w;
    r = fmaf(w, col[k][0], r);
    g = fmaf(w, col[k][1], g);
    b = fmaf(w, col[k][2], b);
  }
  const float delta = fmaxf(__expf((EPSF - zmax) * GAMMA_INV), EPSF);
  const float inv   = 1.0f / (wsum + delta);

  f32x4 o;
  o.x = (r + delta) * inv;   // background = (1,1,1)
  o.y = (g + delta) * inv;
  o.z = (b + delta) * inv;
  o.w = 1.0f - alpha;
  return o;
}

__device__ __forceinline__ void load_pixel_global(PixIn& px, long p,
    const float* __restrict__ bary, const float* __restrict__ dists,
    const float* __restrict__ zbuf, const int* __restrict__ p2f) {
  const f32x4* b4 = (const f32x4*)(bary + p * 24);
#pragma unroll
  for (int i = 0; i < 6; ++i)
    ((f32x4*)px.bary)[i] = __builtin_nontemporal_load(b4 + i);
  const i32x4* f4 = (const i32x4*)(p2f + p * 8);
  ((i32x4*)px.pf)[0] = __builtin_nontemporal_load(f4 + 0);
  ((i32x4*)px.pf)[1] = __builtin_nontemporal_load(f4 + 1);
  const f32x4* d4 = (const f32x4*)(dists + p * 8);
  ((f32x4*)px.dist)[0] = __builtin_nontemporal_load(d4 + 0);
  ((f32x4*)px.dist)[1] = __builtin_nontemporal_load(d4 + 1);
  const f32x4* z4 = (const f32x4*)(zbuf + p * 8);
  ((f32x4*)px.z)[0] = __builtin_nontemporal_load(z4 + 0);
  ((f32x4*)px.z)[1] = __builtin_nontemporal_load(z4 + 1);
}

__global__ __launch_bounds__(TPB) void soft_normal_shader_kernel(
    const float* __restrict__ vn,     // (V,3)
    const float* __restrict__ bary,   // (P,K,3)
    const float* __restrict__ dists,  // (P,K)
    const float* __restrict__ zbuf,   // (P,K)
    const int*   __restrict__ faces,  // (F,3)
    const int*   __restrict__ p2f,    // (P,K)
    float*       __restrict__ out,    // (P,4)
    long P) {
  const int  t    = threadIdx.x;
  const long base = (long)blockIdx.x * TPB;
  const long p    = base + t;

  PixIn px;

#if HAS_ASYNC_LDS
  __shared__ float s_bary[TPB * 24];   // 24 KB
  __shared__ int   s_p2f [TPB * 8];    //  8 KB
  __shared__ float s_dist[TPB * 8];    //  8 KB
  __shared__ float s_z   [TPB * 8];    //  8 KB
  const bool full = (base + TPB) <= P;
  if (full) {
    // Fully coalesced async staging: consecutive lanes copy consecutive 16B.
    const char* gb = (const char*)(bary  + base * 24);
    const char* gf = (const char*)(p2f   + base * 8);
    const char* gd = (const char*)(dists + base * 8);
    const char* gz = (const char*)(zbuf  + base * 8);
#pragma unroll
    for (int i = 0; i < 6; ++i) {
      const int c = t + i * TPB;
      stage16(gb + c * 16, (char*)s_bary + c * 16);
    }
#pragma unroll
    for (int i = 0; i < 2; ++i) {
      const int c = t + i * TPB;
      stage16(gf + c * 16, (char*)s_p2f  + c * 16);
      stage16(gd + c * 16, (char*)s_dist + c * 16);
      stage16(gz + c * 16, (char*)s_z    + c * 16);
    }
    wait_async0();
    __syncthreads();
#pragma unroll
    for (int i = 0; i < 6; ++i)
      ((f32x4*)px.bary)[i] = ((const f32x4*)s_bary)[t * 6 + i];
#pragma unroll
    for (int i = 0; i < 2; ++i) {
      ((i32x4*)px.pf)[i]   = ((const i32x4*)s_p2f )[t * 2 + i];
      ((f32x4*)px.dist)[i] = ((const f32x4*)s_dist)[t * 2 + i];
      ((f32x4*)px.z)[i]    = ((const f32x4*)s_z   )[t * 2 + i];
    }
  } else if (p < P) {
    load_pixel_global(px, p, bary, dists, zbuf, p2f);
  }
  if (p >= P) return;
#else
  if (p >= P) return;
  load_pixel_global(px, p, bary, dists, zbuf, p2f);
#endif

  const f32x4 o = shade(px, vn, faces);
  __builtin_nontemporal_store(o, (f32x4*)out + p);
}

}  // namespace

extern "C" void kernel_launch(void* const* d_in, const int* in_sizes, int n_in,
                              void* d_out, int out_size, void* d_ws, size_t ws_size,
                              hipStream_t stream) {
  (void)n_in; (void)out_size; (void)d_ws; (void)ws_size;
  const float* vn    = (const float*)d_in[0];  // verts_normals (V,3)
  const float* bary  = (const float*)d_in[1];  // bary_coords   (N,H,W,K,3)
  const float* dists = (const float*)d_in[2];  // dists         (N,H,W,K)
  const float* zbuf  = (const float*)d_in[3];  // zbuf          (N,H,W,K)
  const int*   faces = (const int*)d_in[4];    // faces         (F,3)
  const int*   p2f   = (const int*)d_in[5];    // pix_to_face   (N,H,W,K)
  float*       out   = (float*)d_out;          // (N,H,W,4)

  const long P = (long)in_sizes[5] / KF;       // N*H*W pixels
  const dim3 grid((unsigned)((P + TPB - 1) / TPB));
  soft_normal_shader_kernel<<<grid, TPB, 0, stream>>>(
      vn, bary, dists, zbuf, faces, p2f, out, P);
}